// LightGCN_9371618639985
// MI455X (gfx1250) — compile-verified
//
#include <hip/hip_runtime.h>

#define N_USERS 100000
#define N_ITEMS 50000
#define NTOT    150000
#define DD      64
#define NNZ_    10000000
#define BATCH   16384

typedef float v2f __attribute__((ext_vector_type(2)));
typedef float v8f __attribute__((ext_vector_type(8)));

// ---------------- zero fill (float4 vectorized) ----------------
__global__ void zero_kernel(float4* __restrict__ buf, int n4) {
    int i = blockIdx.x * blockDim.x + threadIdx.x;
    if (i < n4) buf[i] = make_float4(0.f, 0.f, 0.f, 0.f);
}

// ---------------- COO SpMM: Eout[r] += v * Ein[c] ----------------
// 16 lanes per edge, float4 per lane -> fully coalesced 256B row access.
__global__ void spmm_kernel(const float* __restrict__ vals,
                            const int*   __restrict__ rows,
                            const int*   __restrict__ cols,
                            const float* __restrict__ Ein,
                            float*       __restrict__ Eout) {
    unsigned t    = blockIdx.x * blockDim.x + threadIdx.x;
    unsigned e    = t >> 4;
    unsigned lane = t & 15u;
    if (e >= NNZ_) return;

    if (lane == 0) {
        // stream-ahead on the COO arrays (emits global_prefetch_b8)
        __builtin_prefetch(vals + e + 4096, 0, 1);
        __builtin_prefetch(rows + e + 4096, 0, 1);
        __builtin_prefetch(cols + e + 4096, 0, 1);
    }

    float v = vals[e];
    int   r = rows[e];
    int   c = cols[e];

    const float4 s = *reinterpret_cast<const float4*>(Ein + (size_t)c * DD + lane * 4);
    float* dst = Eout + (size_t)r * DD + lane * 4;
    atomicAdd(dst + 0, v * s.x);
    atomicAdd(dst + 1, v * s.y);
    atomicAdd(dst + 2, v * s.z);
    atomicAdd(dst + 3, v * s.w);
}

// ---------------- WMMA identity-accumulate: C += I * B_tile ----------------
// A is 16x4 f32: VGPR0 -> K = 0 (lanes 0-15) / K = 2 (lanes 16-31),
//                VGPR1 -> K = 1 / K = 3   (per CDNA5 ISA 7.12.2).
// B is 4x16 f32 with rows lane-striped the same way; C/D 16x16 in 8 VGPRs.
__device__ __forceinline__ v8f tile_add_wmma(v8f c, const float* __restrict__ Eb,
                                             int rowBase, int col, int hi, int lane) {
#pragma unroll
    for (int kk = 0; kk < 4; ++kk) {
        int k0 = kk * 4 + 2 * hi;
        v2f a, b;
        a.x = ((lane & 15) == k0    ) ? 1.0f : 0.0f;
        a.y = ((lane & 15) == k0 + 1) ? 1.0f : 0.0f;
        b.x = Eb[(size_t)(rowBase + k0    ) * DD + col];
        b.y = Eb[(size_t)(rowBase + k0 + 1) * DD + col];
        c = __builtin_amdgcn_wmma_f32_16x16x4_f32(
                /*neg_a=*/false, a, /*neg_b=*/false, b,
                /*c_mod=*/(short)0, c, /*reuse_a=*/false, /*reuse_b=*/false);
    }
    return c;
}

// F = 0.25 * (E0 + E1 + E2 + E3), one wave32 per 16x16 tile.
// NTOT = 150000 = 9375 * 16 exactly; 37500 tiles; 4 waves/block -> 9375 blocks.
__global__ void final_wmma_kernel(const float* __restrict__ E0,
                                  const float* __restrict__ E1,
                                  const float* __restrict__ E2,
                                  const float* __restrict__ E3,
                                  float* F) {
    int wave = blockIdx.x * (blockDim.x >> 5) + (threadIdx.x >> 5);
    int lane = threadIdx.x & 31;
    const int nTiles = (NTOT / 16) * (DD / 16);
    if (wave >= nTiles) return;              // wave-uniform: EXEC stays all-ones

    int rowBase = (wave >> 2) * 16;
    int colBase = (wave & 3) * 16;
    int col = colBase + (lane & 15);
    int hi  = lane >> 4;

    v8f c;
#pragma unroll
    for (int v = 0; v < 8; ++v)
        c[v] = E0[(size_t)(rowBase + v + 8 * hi) * DD + col];

    c = tile_add_wmma(c, E1, rowBase, col, hi, lane);
    c = tile_add_wmma(c, E2, rowBase, col, hi, lane);
    c = tile_add_wmma(c, E3, rowBase, col, hi, lane);

#pragma unroll
    for (int v = 0; v < 8; ++v)
        F[(size_t)(rowBase + v + 8 * hi) * DD + col] = 0.25f * c[v];
}

// ---------------- row gather: out[b,:] = src[rowOffset + idx[b], :] ----------------
__global__ void gather_kernel(const float* __restrict__ src,
                              const int*   __restrict__ idx,
                              int rowOffset, float* __restrict__ out) {
    int t = blockIdx.x * blockDim.x + threadIdx.x;   // BATCH*16 threads (float4 each)
    if (t >= BATCH * 16) return;
    int b = t >> 4, j = t & 15;
    int row = rowOffset + idx[b];
    reinterpret_cast<float4*>(out)[(size_t)b * 16 + j] =
        reinterpret_cast<const float4*>(src)[(size_t)row * 16 + j];
}

extern "C" void kernel_launch(void* const* d_in, const int* in_sizes, int n_in,
                              void* d_out, int out_size, void* d_ws, size_t ws_size,
                              hipStream_t stream) {
    const float* E0   = (const float*)d_in[0];
    const float* vals = (const float*)d_in[1];
    const int*   rows = (const int*)  d_in[2];
    const int*   cols = (const int*)  d_in[3];
    const int*   users = (const int*) d_in[4];
    const int*   pos   = (const int*) d_in[5];
    const int*   neg   = (const int*) d_in[6];
    float* out = (float*)d_out;

    const size_t ND = (size_t)NTOT * DD;
    float* E1 = (float*)d_ws;        // layer 1; later reused for final F
    float* E2 = E1 + ND;             // layer 2
    float* E3 = E2 + ND;             // layer 3

    const int seg = BATCH * DD;

    const int gThreads = BATCH * 16;
    const int gBlock = 256;
    const int gGrid = (gThreads + gBlock - 1) / gBlock;

    // init embedding gathers (outputs 3..5) straight from E0
    gather_kernel<<<gGrid, gBlock, 0, stream>>>(E0, users, 0,       out + 3 * seg);
    gather_kernel<<<gGrid, gBlock, 0, stream>>>(E0, pos,   N_USERS, out + 4 * seg);
    gather_kernel<<<gGrid, gBlock, 0, stream>>>(E0, neg,   N_USERS, out + 5 * seg);

    const int n4    = (int)(ND / 4);
    const int zGrid = (n4 + 255) / 256;
    const int sGrid = (int)(((size_t)NNZ_ * 16 + 255) / 256);   // 625000

    zero_kernel<<<zGrid, 256, 0, stream>>>((float4*)E1, n4);
    spmm_kernel<<<sGrid, 256, 0, stream>>>(vals, rows, cols, E0, E1);

    zero_kernel<<<zGrid, 256, 0, stream>>>((float4*)E2, n4);
    spmm_kernel<<<sGrid, 256, 0, stream>>>(vals, rows, cols, E1, E2);

    zero_kernel<<<zGrid, 256, 0, stream>>>((float4*)E3, n4);
    spmm_kernel<<<sGrid, 256, 0, stream>>>(vals, rows, cols, E2, E3);

    // F = 0.25*(E0+E1+E2+E3) via v_wmma_f32_16x16x4_f32; F written over E1.
    const int nTiles = (NTOT / 16) * (DD / 16);      // 37500
    final_wmma_kernel<<<nTiles / 4, 128, 0, stream>>>(E0, E1, E2, E3, E1);

    // final embedding gathers (outputs 0..2) from F (= E1 buffer)
    gather_kernel<<<gGrid, gBlock, 0, stream>>>(E1, users, 0,       out + 0 * seg);
    gather_kernel<<<gGrid, gBlock, 0, stream>>>(E1, pos,   N_USERS, out + 1 * seg);
    gather_kernel<<<gGrid, gBlock, 0, stream>>>(E1, neg,   N_USERS, out + 2 * seg);
}